// SampleConditionalGMM_4896262718026
// MI455X (gfx1250) — compile-verified
//
#include <hip/hip_runtime.h>
#include <stdint.h>

// ---------------------------------------------------------------------------
// SampleConditionalGMM for MI455X (gfx1250).
//
// out[b,v,c] = stds[b, lut[label[b,v]], c] * noise[b,v,c]
//            + means[b, lut[label[b,v]], c]
//
// Pure HBM-streaming problem: ~164 MB moved once => ~7 us at 23.3 TB/s.
// No matrix structure -> no WMMA. Strategy:
//   * Fold the label->row LUT into an LDS table indexed by raw label value:
//       s_tbl[b*48 + label] = (mean_c0, mean_c1, std_c0, std_c1)   (float4)
//   * Stage that 1.5 KB table with CDNA5 async global->LDS loads
//     (global_load_async_to_lds_b64, ASYNCcnt, s_wait_asynccnt).
//   * Hot loop: 4 voxels/thread, all 128-bit vmem with NT temporal hints
//     (single-pass data, keep L2 clean), ds_load_b128 table gathers, FMAs.
// ---------------------------------------------------------------------------

// Native clang vectors (required by __builtin_nontemporal_*; HIP's
// float4/int4 are structs and are rejected).
typedef float vf4 __attribute__((ext_vector_type(4)));
typedef int   vi4 __attribute__((ext_vector_type(4)));

namespace {
constexpr int kNLabels      = 25;
constexpr int kMaxLabel     = 48;                     // table rows per batch
constexpr int kCh           = 2;
constexpr int kBatch        = 2;
constexpr int kDHW          = 160 * 160 * 160;        // 4,096,000 voxels/batch
constexpr int kTotalVox     = kBatch * kDHW;          // 8,192,000
constexpr int kVoxPerThread = 4;
constexpr int kThreads      = kTotalVox / kVoxPerThread; // 2,048,000
constexpr int kBlock        = 256;                    // 8 waves (wave32)
constexpr int kGrid         = kThreads / kBlock;      // 8,000 (exact)
constexpr int kFillOps      = kBatch * kNLabels * 2;  // 100 async b64 ops
}  // namespace

__device__ __constant__ int c_gen_labels[kNLabels] = {
    0, 2, 3, 4, 5, 7, 8, 10, 11, 12, 13, 14, 15, 16, 17, 18,
    24, 26, 28, 41, 42, 43, 44, 46, 47};

__global__ __launch_bounds__(kBlock) void gmm_sample_kernel(
    const int*   __restrict__ labels,   // [B*DHW]        (trailing dim 1 dropped)
    const float* __restrict__ means,    // [B, 25, 2]
    const float* __restrict__ stds,     // [B, 25, 2]
    const float* __restrict__ noise,    // [B*DHW, 2]
    float*       __restrict__ out)      // [B*DHW, 2]
{
  // Per-(batch,label) entry: {mean_c0, mean_c1, std_c0, std_c1}
  __shared__ vf4 s_tbl[kBatch * kMaxLabel];   // 96 * 16B = 1536 B

  const int t = threadIdx.x;

  // Deterministic init of rows whose label value never occurs.
  if (t < kBatch * kMaxLabel) {
    s_tbl[t] = (vf4)0.f;
  }
  __syncthreads();

  // ---- Stage table via CDNA5 async global->LDS scatter (ASYNCcnt path) ----
  // 100 lanes: lane (b, l, q) copies 8 bytes (two channel values) of
  // means (q=0) or stds (q=1) for GMM row l into LDS row GEN_LABELS[l].
  if (t < kFillOps) {
    const int q  = t & 1;                 // 0 = mean pair, 1 = std pair
    const int bl = t >> 1;                // 0..49
    const int b  = bl / kNLabels;
    const int l  = bl % kNLabels;

    const float* src = (q ? stds : means) + (b * kNLabels + l) * kCh;
    const unsigned lds_base = (unsigned)(uintptr_t)(&s_tbl[0]);
    const unsigned lds_byte =
        lds_base + (unsigned)((b * kMaxLabel + c_gen_labels[l]) * 16 + q * 8);

    asm volatile("global_load_async_to_lds_b64 %0, %1, off"
                 :
                 : "v"(lds_byte), "v"((uint64_t)(uintptr_t)src)
                 : "memory");
  }
  asm volatile("s_wait_asynccnt 0" ::: "memory");
  __syncthreads();

  // ------------------------------ Hot loop --------------------------------
  const int tid = blockIdx.x * kBlock + t;      // 0 .. 2,047,999 (exact grid)
  const int v0  = tid * kVoxPerThread;          // first voxel, multiple of 4
  const int boff = (v0 >= kDHW) ? kMaxLabel : 0;  // batch row offset (B == 2)

  // Single-pass data: non-temporal (th:NT) 128-bit loads/stores.
  const vi4 lab4 = __builtin_nontemporal_load((const vi4*)(labels + v0));
  const vf4 n0 = __builtin_nontemporal_load((const vf4*)(noise + v0 * kCh));
  const vf4 n1 = __builtin_nontemporal_load((const vf4*)(noise + v0 * kCh + 4));

  const vf4 e0 = s_tbl[boff + lab4.x];          // ds_load_b128
  const vf4 e1 = s_tbl[boff + lab4.y];
  const vf4 e2 = s_tbl[boff + lab4.z];
  const vf4 e3 = s_tbl[boff + lab4.w];

  vf4 r0, r1;
  r0.x = fmaf(e0.z, n0.x, e0.x);   // voxel0 c0: std0*n + mean0
  r0.y = fmaf(e0.w, n0.y, e0.y);   // voxel0 c1
  r0.z = fmaf(e1.z, n0.z, e1.x);   // voxel1 c0
  r0.w = fmaf(e1.w, n0.w, e1.y);   // voxel1 c1
  r1.x = fmaf(e2.z, n1.x, e2.x);   // voxel2 c0
  r1.y = fmaf(e2.w, n1.y, e2.y);   // voxel2 c1
  r1.z = fmaf(e3.z, n1.z, e3.x);   // voxel3 c0
  r1.w = fmaf(e3.w, n1.w, e3.y);   // voxel3 c1

  __builtin_nontemporal_store(r0, (vf4*)(out + v0 * kCh));
  __builtin_nontemporal_store(r1, (vf4*)(out + v0 * kCh + 4));
}

extern "C" void kernel_launch(void* const* d_in, const int* in_sizes, int n_in,
                              void* d_out, int out_size, void* d_ws, size_t ws_size,
                              hipStream_t stream) {
  (void)in_sizes; (void)n_in; (void)out_size; (void)d_ws; (void)ws_size;
  const int*   labels = (const int*)  d_in[0];   // label_map [2,160,160,160,1] i32
  const float* means  = (const float*)d_in[1];   // [2,25,2] f32
  const float* stds   = (const float*)d_in[2];   // [2,25,2] f32
  const float* noise  = (const float*)d_in[3];   // [2,160,160,160,2] f32
  float*       out    = (float*)d_out;           // [2,160,160,160,2] f32

  gmm_sample_kernel<<<dim3(kGrid), dim3(kBlock), 0, stream>>>(
      labels, means, stds, noise, out);
}